// HGNNExpertCoupler_7060926234651
// MI455X (gfx1250) — compile-verified
//
#include <hip/hip_runtime.h>
#include <hip/hip_bf16.h>
#include <math.h>

// ---- CDNA5 WMMA types ----
typedef __attribute__((ext_vector_type(2))) float v2f;
typedef __attribute__((ext_vector_type(8))) float v8f;

#define DM 512
#define XSTR 516          // LDS row stride (floats): 516 % 64 == 4 -> conflict-free b64 A-frag loads
#define EPS 1e-5f

__device__ __forceinline__ float gelu_exact(float x) {
    return 0.5f * x * (1.0f + erff(x * 0.70710678118654752440f));
}

// C = A @ B, all 512x512 row-major (tiny weight-folding GEMM; perf-irrelevant)
__global__ void fold_gemm512(const float* __restrict__ A,
                             const float* __restrict__ B,
                             float* __restrict__ C) {
    int n = blockIdx.x * blockDim.x + threadIdx.x;   // 0..262143
    int r = n >> 9, c = n & (DM - 1);
    const float* __restrict__ ar = A + (size_t)r * DM;
    float s = 0.f;
    #pragma unroll 8
    for (int k = 0; k < DM; ++k) s = fmaf(ar[k], B[(size_t)k * DM + c], s);
    C[n] = s;
}

// t[k] = b1 @ W2 + b2
__global__ void fold_bias1(const float* __restrict__ b1, const float* __restrict__ W2,
                           const float* __restrict__ b2, float* __restrict__ t) {
    int k = blockIdx.x * blockDim.x + threadIdx.x;
    if (k >= DM) return;
    float s = b2[k];
    #pragma unroll 8
    for (int j = 0; j < DM; ++j) s = fmaf(b1[j], W2[(size_t)j * DM + k], s);
    t[k] = s;
}

// beff[n] = t @ Wc + bc
__global__ void fold_bias2(const float* __restrict__ t, const float* __restrict__ Wc,
                           const float* __restrict__ bc, float* __restrict__ beff) {
    int n = blockIdx.x * blockDim.x + threadIdx.x;
    if (n >= DM) return;
    float s = bc[n];
    #pragma unroll 8
    for (int k = 0; k < DM; ++k) s = fmaf(t[k], Wc[(size_t)k * DM + n], s);
    beff[n] = s;
}

// Fused: expert-mean -> [16x512]x[512x512] fp32 WMMA GEMM -> exact GELU -> LayerNorm
// grid.x = 16384/16 = 1024 blocks, 128 threads (4 wave32)
__global__ __launch_bounds__(128) void hgnn_fused(
    const float* __restrict__ x,      // [16384, 8, 512] (B*L flattened)
    const float* __restrict__ Weff,   // [512, 512] row-major (k, n)
    const float* __restrict__ beff,   // [512]
    const float* __restrict__ gamma,  // [512]
    const float* __restrict__ beta,   // [512]
    float* __restrict__ out)          // [16384, 512]
{
    __shared__ float s_x[16 * XSTR];      // xbar tile, then reused for y tile
    __shared__ float s_ps[16 * 8];
    __shared__ float s_pss[16 * 8];
    __shared__ float s_mu[16];
    __shared__ float s_rstd[16];

    const int tid  = threadIdx.x;
    const int lane = tid & 31;
    const int wv   = tid >> 5;            // wave id 0..3
    const size_t row0 = (size_t)blockIdx.x * 16;

    // ---------- Phase 1: mean over experts -> LDS xbar tile [16][512] ----------
    // Each block streams a contiguous 256KB chunk of expert_outputs.
    for (int idx = tid; idx < 16 * 128; idx += 128) {
        const int r  = idx >> 7;          // tile row 0..15
        const int c4 = idx & 127;         // float4 column 0..127
        float4 acc = make_float4(0.f, 0.f, 0.f, 0.f);
        const float* base = x + ((row0 + r) * 8) * DM + c4 * 4;
        #pragma unroll
        for (int e = 0; e < 8; ++e) {
            const float4 p = *(const float4*)(base + (size_t)e * DM);
            acc.x += p.x; acc.y += p.y; acc.z += p.z; acc.w += p.w;
        }
        acc.x *= 0.125f; acc.y *= 0.125f; acc.z *= 0.125f; acc.w *= 0.125f;
        *(float4*)&s_x[r * XSTR + c4 * 4] = acc;   // 2064B row stride: 16B aligned
    }
    __syncthreads();

    // ---------- Phase 2: y = xbar @ Weff via V_WMMA_F32_16X16X4_F32 ----------
    // Lane roles per ISA layout:
    //   A frag (16x4 fp32): lane holds A[lane&15][2*(lane>>4) + v], v=0..1
    //   B frag (4x16 fp32): lane holds B[2*(lane>>4) + v][lane&15]
    //   C/D (16x16 fp32):   vgpr j, lane -> D[(lane>>4)*8 + j][lane&15]
    const int col = lane & 15;
    const int kg  = lane >> 4;            // 0 or 1
    const int ncolbase = wv * 128 + col;  // this wave owns global cols [wv*128, wv*128+128)

    v8f acc[8] = {};                      // 8 N-tiles of 16 cols each

    for (int k0 = 0; k0 < DM; k0 += 4) {
        const int krow = k0 + 2 * kg;
        // A fragment from LDS (conflict-free: 4*r+{0..3} covers all 64 banks)
        const v2f a = *(const v2f*)&s_x[col * XSTR + krow];
        const float* __restrict__ wrow = Weff + (size_t)krow * DM + ncolbase;
        #pragma unroll
        for (int nt = 0; nt < 8; ++nt) {
            v2f b;
            b.x = wrow[nt * 16];
            b.y = wrow[DM + nt * 16];
            acc[nt] = __builtin_amdgcn_wmma_f32_16x16x4_f32(
                false, a, false, b, (short)0, acc[nt], false, false);
        }
    }
    __syncthreads();   // everyone done reading xbar; LDS is reused for y

    // ---------- Phase 3a: bias + exact GELU -> LDS y tile ----------
    #pragma unroll
    for (int nt = 0; nt < 8; ++nt) {
        const int c = ncolbase + nt * 16;
        const float badd = beff[c];
        #pragma unroll
        for (int j = 0; j < 8; ++j) {
            const int r = kg * 8 + j;
            s_x[r * XSTR + c] = gelu_exact(acc[nt][j] + badd);
        }
    }
    __syncthreads();

    // ---------- Phase 3b: LayerNorm over 512 per row ----------
    {
        const int r = tid >> 3, sub = tid & 7;     // 8 threads per row, 64 cols each
        const float* yr = &s_x[r * XSTR + sub * 64];
        float s = 0.f, ss = 0.f;
        #pragma unroll 8
        for (int i = 0; i < 64; ++i) { const float v = yr[i]; s += v; ss = fmaf(v, v, ss); }
        s_ps[r * 8 + sub]  = s;
        s_pss[r * 8 + sub] = ss;
    }
    __syncthreads();
    if ((tid & 7) == 0) {
        const int r = tid >> 3;
        float ts = 0.f, tss = 0.f;
        #pragma unroll
        for (int i = 0; i < 8; ++i) { ts += s_ps[r * 8 + i]; tss += s_pss[r * 8 + i]; }
        const float mu  = ts * (1.f / 512.f);
        const float var = tss * (1.f / 512.f) - mu * mu;   // biased var, matches jnp.var
        s_mu[r]   = mu;
        s_rstd[r] = rsqrtf(var + EPS);
    }
    __syncthreads();

    // Coalesced normalized store
    for (int idx = tid; idx < 16 * DM; idx += 128) {
        const int r = idx >> 9, c = idx & (DM - 1);
        const float v = (s_x[r * XSTR + c] - s_mu[r]) * s_rstd[r];
        out[(row0 + r) * DM + c] = fmaf(v, gamma[c], beta[c]);
    }
}

extern "C" void kernel_launch(void* const* d_in, const int* in_sizes, int n_in,
                              void* d_out, int out_size, void* d_ws, size_t ws_size,
                              hipStream_t stream) {
    const float* x     = (const float*)d_in[0]; // [8,2048,8,512]
    const float* W1    = (const float*)d_in[1];
    const float* b1    = (const float*)d_in[2];
    const float* W2    = (const float*)d_in[3];
    const float* b2    = (const float*)d_in[4];
    const float* Wc    = (const float*)d_in[5];
    const float* bc    = (const float*)d_in[6];
    const float* gamma = (const float*)d_in[7];
    const float* beta  = (const float*)d_in[8];
    float* out = (float*)d_out;

    float* ws   = (float*)d_ws;              // needs ~2.1 MB
    float* Wtmp = ws;                        // 512*512
    float* Weff = ws + DM * DM;              // 512*512
    float* tvec = ws + 2 * DM * DM;          // 512
    float* beff = tvec + DM;                 // 512

    // Fold weights: Weff = W1 @ W2 @ Wc ; beff = (b1 @ W2 + b2) @ Wc + bc
    fold_gemm512<<<(DM * DM) / 256, 256, 0, stream>>>(W1, W2, Wtmp);
    fold_gemm512<<<(DM * DM) / 256, 256, 0, stream>>>(Wtmp, Wc, Weff);
    fold_bias1<<<2, 256, 0, stream>>>(b1, W2, b2, tvec);
    fold_bias2<<<2, 256, 0, stream>>>(tvec, Wc, bc, beff);

    // Fused streaming kernel: 16384 rows / 16 per block
    hgnn_fused<<<1024, 128, 0, stream>>>(x, Weff, beff, gamma, beta, out);
}